// CGGRUForceStress_37194416783625
// MI455X (gfx1250) — compile-verified
//
#include <hip/hip_runtime.h>
#include <math.h>

typedef __attribute__((ext_vector_type(2))) float v2f;
typedef __attribute__((ext_vector_type(8))) float v8f;

#define NNODES 50000
#define NEDGES 1000000
#define DDIM 64
#define FIN 19
#define EHD 50
#define BGR 128
#define NBLK 3
#define PI_F 3.14159265358979323846f
#define LOG2_F 0.69314718055994531f

__device__ inline float sigmoidf_(float x) { return 1.f / (1.f + expf(-x)); }

__device__ inline v8f vzero8() {
  v8f c = {0.f, 0.f, 0.f, 0.f, 0.f, 0.f, 0.f, 0.f};
  return c;
}

// D(16x16) += A(16x4) * B(4x16), f32 WMMA, accumulate in C
__device__ inline v8f wmma4(v2f a, v2f b, v8f c) {
  return __builtin_amdgcn_wmma_f32_16x16x4_f32(false, a, false, b, (short)0, c, false, false);
}

__global__ void k_fill(float* p, float v, int n) {
  int i = blockIdx.x * blockDim.x + threadIdx.x;
  if (i < n) p[i] = v;
}

__global__ void k_env_deg(const float* __restrict__ ew, const int* __restrict__ ei,
                          float* __restrict__ cenv, float* __restrict__ deg) {
  int e = blockIdx.x * blockDim.x + threadIdx.x;
  if (e < NEDGES) {
    cenv[e] = cosf(ew[e] * PI_F / 5.0f) * 0.5f + 0.5f;
    atomicAdd(&deg[ei[NEDGES + e]], 1.0f);
  }
}

__global__ void k_lin0(const float* __restrict__ x, const float* __restrict__ w,
                       const float* __restrict__ b, float* __restrict__ h) {
  int i = blockIdx.x * blockDim.x + threadIdx.x;
  if (i < NNODES * DDIM) {
    int n = i >> 6, d = i & 63;
    float acc = b[d];
    for (int k = 0; k < FIN; k++) acc += x[n * FIN + k] * w[d * FIN + k];
    h[i] = acc > 0.f ? acc : 0.01f * acc;  // leaky_relu, slope 0.01
  }
}

// hx = h @ W^T   (W: 64x64 row-major), one 16-row tile per wave
__global__ void __launch_bounds__(256) k_hx(const float* __restrict__ h,
                                            const float* __restrict__ w,
                                            float* __restrict__ hx) {
  __shared__ float s_at[8][16 * 64];
  int wid = threadIdx.x >> 5, lane = threadIdx.x & 31;
  int tile = blockIdx.x * 8 + wid;
  if (tile >= NNODES / 16) return;
  float* at = s_at[wid];
  const float* gsrc = h + (size_t)tile * 1024;
  for (int i = lane; i < 1024; i += 32) at[i] = gsrc[i];
  int half = lane >> 4, m = lane & 15;
  for (int t = 0; t < 4; t++) {
    int col = t * 16 + m;
    v8f c = vzero8();
    for (int kk = 0; kk < 16; kk++) {
      int kb = kk * 4 + half * 2;
      v2f a, bb;
      a.x = at[m * 64 + kb]; a.y = at[m * 64 + kb + 1];
      bb.x = w[col * 64 + kb]; bb.y = w[col * 64 + kb + 1];
      c = wmma4(a, bb, c);
    }
    int rb = tile * 16 + half * 8;
    for (int v = 0; v < 8; v++) hx[(size_t)(rb + v) * 64 + col] = c[v];
  }
}

// Per-edge: Wm = (relu(ea@w1^T+b1)@w2^T+b2)*Cenv ; msg = hx[src]*Wm ; agg[dst] += msg
__global__ void __launch_bounds__(256) k_edge(
    const float* __restrict__ ea, const int* __restrict__ ei,
    const float* __restrict__ cenv, const float* __restrict__ hx,
    const float* __restrict__ w1, const float* __restrict__ b1,
    const float* __restrict__ w2, const float* __restrict__ b2,
    float* __restrict__ agg) {
  __shared__ float s_at[8][16 * 52];
  __shared__ float s_mid[8][16 * 32];
  __shared__ int s_src[8][16];
  __shared__ int s_dst[8][16];
  __shared__ float s_ce[8][16];
  int wid = threadIdx.x >> 5, lane = threadIdx.x & 31;
  int tile = blockIdx.x * 8 + wid;
  if (tile >= NEDGES / 16) return;
  int eb = tile * 16;
  float* at = s_at[wid];
  float* mid = s_mid[wid];
  if (lane < 16) {
    s_src[wid][lane] = ei[eb + lane];
    s_ce[wid][lane] = cenv[eb + lane];
  } else {
    s_dst[wid][lane - 16] = ei[NEDGES + eb + lane - 16];
  }
  for (int i = lane; i < 16 * 52; i += 32) {
    int r = i / 52, cc = i % 52;
    at[i] = cc < EHD ? ea[(size_t)(eb + r) * EHD + cc] : 0.f;
  }
  int half = lane >> 4, m = lane & 15;
  // GEMM1: (16x52) x (52x32), K padded with zeros; w1 is 32x50 row-major
  for (int t = 0; t < 2; t++) {
    int col = t * 16 + m;
    v8f c = vzero8();
    for (int kk = 0; kk < 13; kk++) {
      int kb = kk * 4 + half * 2;
      v2f a, bb;
      a.x = at[m * 52 + kb]; a.y = at[m * 52 + kb + 1];
      bb.x = (kb < EHD) ? w1[col * EHD + kb] : 0.f;
      bb.y = (kb + 1 < EHD) ? w1[col * EHD + kb + 1] : 0.f;
      c = wmma4(a, bb, c);
    }
    float bv = b1[col];
    for (int v = 0; v < 8; v++) {
      float t1 = c[v] + bv;
      mid[(v + 8 * half) * 32 + col] = t1 > 0.f ? t1 : 0.f;  // ReLU
    }
  }
  // GEMM2: (16x32) x (32x64); w2 is 64x32 row-major
  for (int t = 0; t < 4; t++) {
    int col = t * 16 + m;
    v8f c = vzero8();
    for (int kk = 0; kk < 8; kk++) {
      int kb = kk * 4 + half * 2;
      v2f a, bb;
      a.x = mid[m * 32 + kb]; a.y = mid[m * 32 + kb + 1];
      bb.x = w2[col * 32 + kb]; bb.y = w2[col * 32 + kb + 1];
      c = wmma4(a, bb, c);
    }
    float bv = b2[col];
    for (int v = 0; v < 8; v++) {
      int row = v + 8 * half;
      float wm = (c[v] + bv) * s_ce[wid][row];
      float msg = wm * hx[(size_t)s_src[wid][row] * DDIM + col];
      atomicAdd(&agg[(size_t)s_dst[wid][row] * DDIM + col], msg);
    }
  }
}

// m = ssp((agg/deg)@lin2^T + b2) @ lin3^T + b3
__global__ void __launch_bounds__(256) k_m(
    const float* __restrict__ agg, const float* __restrict__ deg,
    const float* __restrict__ w2, const float* __restrict__ b2,
    const float* __restrict__ w3, const float* __restrict__ b3,
    float* __restrict__ mout) {
  __shared__ float s_at[8][16 * 64];
  int wid = threadIdx.x >> 5, lane = threadIdx.x & 31;
  int tile = blockIdx.x * 8 + wid;
  if (tile >= NNODES / 16) return;
  float* at = s_at[wid];
  int nb = tile * 16;
  for (int i = lane; i < 1024; i += 32) {
    int r = i >> 6;
    float dv = deg[nb + r];
    dv = dv < 1.f ? 1.f : dv;
    at[i] = agg[(size_t)nb * 64 + i] / dv;
  }
  int half = lane >> 4, m = lane & 15;
  v8f cs[4];
  for (int t = 0; t < 4; t++) {
    int col = t * 16 + m;
    v8f c = vzero8();
    for (int kk = 0; kk < 16; kk++) {
      int kb = kk * 4 + half * 2;
      v2f a, bb;
      a.x = at[m * 64 + kb]; a.y = at[m * 64 + kb + 1];
      bb.x = w2[col * 64 + kb]; bb.y = w2[col * 64 + kb + 1];
      c = wmma4(a, bb, c);
    }
    cs[t] = c;
  }
  // overwrite A tile with ssp(t1) (all GEMM1 reads are complete in program order)
  for (int t = 0; t < 4; t++) {
    int col = t * 16 + m;
    float bv = b2[col];
    for (int v = 0; v < 8; v++) {
      float xx = cs[t][v] + bv;
      float sp = fmaxf(xx, 0.f) + log1pf(expf(-fabsf(xx)));
      at[(v + 8 * half) * 64 + col] = sp - LOG2_F;
    }
  }
  for (int t = 0; t < 4; t++) {
    int col = t * 16 + m;
    v8f c = vzero8();
    for (int kk = 0; kk < 16; kk++) {
      int kb = kk * 4 + half * 2;
      v2f a, bb;
      a.x = at[m * 64 + kb]; a.y = at[m * 64 + kb + 1];
      bb.x = w3[col * 64 + kb]; bb.y = w3[col * 64 + kb + 1];
      c = wmma4(a, bb, c);
    }
    float bv = b3[col];
    int rb = nb + half * 8;
    for (int v = 0; v < 8; v++) mout[(size_t)(rb + v) * 64 + col] = c[v] + bv;
  }
}

// GRU step, fused: gi/gh fragments via WMMA, elementwise on fragments, h updated in place
__global__ void __launch_bounds__(128) k_gru(
    const float* __restrict__ mbuf, float* __restrict__ h,
    const float* __restrict__ wih, const float* __restrict__ whh,
    const float* __restrict__ bih, const float* __restrict__ bhh) {
  __shared__ float s_m[4][1024];
  __shared__ float s_h[4][1024];
  int wid = threadIdx.x >> 5, lane = threadIdx.x & 31;
  int tile = blockIdx.x * 4 + wid;
  if (tile >= NNODES / 16) return;
  float* mt = s_m[wid];
  float* ht = s_h[wid];
  int nb = tile * 16;
  for (int i = lane; i < 1024; i += 32) {
    mt[i] = mbuf[(size_t)nb * 64 + i];
    ht[i] = h[(size_t)nb * 64 + i];
  }
  int half = lane >> 4, m = lane & 15;
  for (int t = 0; t < 4; t++) {
    int c0 = t * 16 + m, c1 = 64 + c0, c2 = 128 + c0;
    v8f cir = vzero8(), ciz = vzero8(), cin = vzero8();
    v8f chr_ = vzero8(), chz = vzero8(), chn = vzero8();
    for (int kk = 0; kk < 16; kk++) {
      int kb = kk * 4 + half * 2;
      v2f am, ah, b0, b1v, b2v, b3v, b4v, b5v;
      am.x = mt[m * 64 + kb]; am.y = mt[m * 64 + kb + 1];
      ah.x = ht[m * 64 + kb]; ah.y = ht[m * 64 + kb + 1];
      b0.x = wih[c0 * 64 + kb]; b0.y = wih[c0 * 64 + kb + 1];
      b1v.x = wih[c1 * 64 + kb]; b1v.y = wih[c1 * 64 + kb + 1];
      b2v.x = wih[c2 * 64 + kb]; b2v.y = wih[c2 * 64 + kb + 1];
      b3v.x = whh[c0 * 64 + kb]; b3v.y = whh[c0 * 64 + kb + 1];
      b4v.x = whh[c1 * 64 + kb]; b4v.y = whh[c1 * 64 + kb + 1];
      b5v.x = whh[c2 * 64 + kb]; b5v.y = whh[c2 * 64 + kb + 1];
      cir = wmma4(am, b0, cir); ciz = wmma4(am, b1v, ciz); cin = wmma4(am, b2v, cin);
      chr_ = wmma4(ah, b3v, chr_); chz = wmma4(ah, b4v, chz); chn = wmma4(ah, b5v, chn);
    }
    float bi0 = bih[c0], bi1 = bih[c1], bi2 = bih[c2];
    float bh0 = bhh[c0], bh1 = bhh[c1], bh2 = bhh[c2];
    for (int v = 0; v < 8; v++) {
      int row = v + 8 * half;
      float r = sigmoidf_(cir[v] + bi0 + chr_[v] + bh0);
      float z = sigmoidf_(ciz[v] + bi1 + chz[v] + bh1);
      float n = tanhf(cin[v] + bi2 + r * (chn[v] + bh2));
      float hold = ht[row * 64 + c0];
      h[(size_t)(nb + row) * 64 + c0] = (1.f - z) * n + z * hold;
    }
  }
}

__global__ void k_lstm_gates(const float* __restrict__ qstar, const float* __restrict__ hl,
                             const float* __restrict__ wih, const float* __restrict__ whh,
                             const float* __restrict__ bih, const float* __restrict__ bhh,
                             float* __restrict__ gates) {
  int i = blockIdx.x * blockDim.x + threadIdx.x;
  if (i < BGR * 256) {
    int g = i >> 8, c = i & 255;
    float acc = bih[c] + bhh[c];
    for (int j = 0; j < 128; j++) acc += qstar[g * 128 + j] * wih[c * 128 + j];
    for (int j = 0; j < 64; j++) acc += hl[g * 64 + j] * whh[c * 64 + j];
    gates[i] = acc;
  }
}

__global__ void k_lstm_update(const float* __restrict__ gates, float* cl, float* hl) {
  int i = blockIdx.x * blockDim.x + threadIdx.x;
  if (i < BGR * 64) {
    int g = i >> 6, d = i & 63;
    const float* gr = gates + g * 256;
    float ig = sigmoidf_(gr[d]), fg = sigmoidf_(gr[64 + d]);
    float gg = tanhf(gr[128 + d]), og = sigmoidf_(gr[192 + d]);
    float c = fg * cl[i] + ig * gg;
    cl[i] = c;
    hl[i] = og * tanhf(c);
  }
}

__global__ void k_attn_e(const float* __restrict__ h, const float* __restrict__ hl,
                         const int* __restrict__ batch, float* __restrict__ esc,
                         float* __restrict__ emax) {
  int n = blockIdx.x * blockDim.x + threadIdx.x;
  if (n < NNODES) {
    int g = batch[n];
    float acc = 0.f;
    for (int d = 0; d < 64; d++) acc += h[(size_t)n * 64 + d] * hl[g * 64 + d];
    esc[n] = acc;
    if (acc >= 0.f) atomicMax((int*)&emax[g], __float_as_int(acc));
    else atomicMin((unsigned int*)&emax[g], __float_as_uint(acc));
  }
}

__global__ void k_attn_a(const float* __restrict__ esc, const float* __restrict__ emax,
                         const int* __restrict__ batch, float* __restrict__ asc,
                         float* __restrict__ asum) {
  int n = blockIdx.x * blockDim.x + threadIdx.x;
  if (n < NNODES) {
    int g = batch[n];
    float a = expf(esc[n] - emax[g]);
    asc[n] = a;
    atomicAdd(&asum[g], a);
  }
}

__global__ void k_readout(const float* __restrict__ asc, const float* __restrict__ asum,
                          const float* __restrict__ h, const int* __restrict__ batch,
                          float* __restrict__ rread) {
  int i = blockIdx.x * blockDim.x + threadIdx.x;
  if (i < NNODES * 64) {
    int n = i >> 6, d = i & 63;
    int g = batch[n];
    atomicAdd(&rread[g * 64 + d], (asc[n] / asum[g]) * h[i]);
  }
}

__global__ void k_qstar(const float* __restrict__ hl, const float* __restrict__ rread,
                        float* __restrict__ qstar) {
  int i = blockIdx.x * blockDim.x + threadIdx.x;
  if (i < BGR * 128) {
    int g = i >> 7, j = i & 127;
    qstar[i] = j < 64 ? hl[g * 64 + j] : rread[g * 64 + j - 64];
  }
}

__global__ void k_glin(const float* __restrict__ qstar, const float* __restrict__ w,
                       const float* __restrict__ b, float* __restrict__ gbuf) {
  int i = blockIdx.x * blockDim.x + threadIdx.x;
  if (i < BGR * 64) {
    int g = i >> 6, d = i & 63;
    float acc = b[d];
    for (int j = 0; j < 128; j++) acc += qstar[g * 128 + j] * w[d * 128 + j];
    gbuf[i] = acc > 0.f ? acc : 0.f;
  }
}

__global__ void k_ref(const int* __restrict__ z, const int* __restrict__ batch,
                      const float* __restrict__ aref, const float* __restrict__ eref,
                      float* __restrict__ refb) {
  int n = blockIdx.x * blockDim.x + threadIdx.x;
  if (n < NNODES) {
    int zz = z[n];
    atomicAdd(&refb[batch[n]], eref[zz] + aref[zz]);
  }
}

__global__ void k_energy(const float* __restrict__ gbuf, const float* __restrict__ w,
                         const float* __restrict__ b, const float* __restrict__ refb,
                         float* __restrict__ out) {
  int g = blockIdx.x * blockDim.x + threadIdx.x;
  if (g < BGR) {
    float acc = b[0];
    for (int d = 0; d < 64; d++) acc += gbuf[g * 64 + d] * w[d];
    out[g] = acc + refb[g];
  }
}

extern "C" void kernel_launch(void* const* d_in, const int* in_sizes, int n_in,
                              void* d_out, int out_size, void* d_ws, size_t ws_size,
                              hipStream_t stream) {
  const float* x = (const float*)d_in[0];
  const int* ei = (const int*)d_in[1];
  const float* ew = (const float*)d_in[2];
  const float* ea = (const float*)d_in[3];
  const int* z = (const int*)d_in[4];
  const int* batch = (const int*)d_in[5];
  const float* lin0_w = (const float*)d_in[6];
  const float* lin0_b = (const float*)d_in[7];
  const float* cw1 = (const float*)d_in[8];
  const float* cb1 = (const float*)d_in[9];
  const float* cw2 = (const float*)d_in[10];
  const float* cb2 = (const float*)d_in[11];
  const float* cl1 = (const float*)d_in[12];
  const float* cl2w = (const float*)d_in[13];
  const float* cl2b = (const float*)d_in[14];
  const float* cl3w = (const float*)d_in[15];
  const float* cl3b = (const float*)d_in[16];
  const float* gwih = (const float*)d_in[17];
  const float* gwhh = (const float*)d_in[18];
  const float* gbih = (const float*)d_in[19];
  const float* gbhh = (const float*)d_in[20];
  const float* lwih = (const float*)d_in[21];
  const float* lwhh = (const float*)d_in[22];
  const float* lbih = (const float*)d_in[23];
  const float* lbhh = (const float*)d_in[24];
  const float* l1w = (const float*)d_in[25];
  const float* l1b = (const float*)d_in[26];
  const float* l2w = (const float*)d_in[27];
  const float* l2b = (const float*)d_in[28];
  const float* aref = (const float*)d_in[29];
  const float* eref = (const float*)d_in[30];
  float* out = (float*)d_out;

  float* W = (float*)d_ws;
  size_t off = 0;
  auto alloc = [&](size_t n) { float* p = W + off; off += n; return p; };
  float* h = alloc((size_t)NNODES * 64);
  float* hx = alloc((size_t)NNODES * 64);
  float* agg = alloc((size_t)NNODES * 64);
  float* mbuf = alloc((size_t)NNODES * 64);
  float* deg = alloc(NNODES);
  float* cenv = alloc(NEDGES);
  float* esc = alloc(NNODES);
  float* asc = alloc(NNODES);
  float* emax = alloc(BGR);
  float* asum = alloc(BGR);
  float* rread = alloc(BGR * 64);
  float* qstar = alloc(BGR * 128);
  float* hl = alloc(BGR * 64);
  float* clb = alloc(BGR * 64);
  float* gates = alloc(BGR * 256);
  float* gbuf = alloc(BGR * 64);
  float* refb = alloc(BGR);
  (void)ws_size; (void)in_sizes; (void)n_in; (void)out_size;

  k_fill<<<(NNODES + 255) / 256, 256, 0, stream>>>(deg, 0.f, NNODES);
  k_env_deg<<<(NEDGES + 255) / 256, 256, 0, stream>>>(ew, ei, cenv, deg);
  k_lin0<<<(NNODES * 64 + 255) / 256, 256, 0, stream>>>(x, lin0_w, lin0_b, h);

  for (int b = 0; b < NBLK; b++) {
    k_fill<<<(NNODES * 64 + 255) / 256, 256, 0, stream>>>(agg, 0.f, NNODES * 64);
    k_hx<<<(NNODES / 16 + 7) / 8, 256, 0, stream>>>(h, cl1 + b * 64 * 64, hx);
    k_edge<<<(NEDGES / 16 + 7) / 8, 256, 0, stream>>>(
        ea, ei, cenv, hx, cw1 + b * 32 * EHD, cb1 + b * 32, cw2 + b * 64 * 32,
        cb2 + b * 64, agg);
    k_m<<<(NNODES / 16 + 7) / 8, 256, 0, stream>>>(
        agg, deg, cl2w + b * 64 * 64, cl2b + b * 64, cl3w + b * 64 * 64,
        cl3b + b * 64, mbuf);
    k_gru<<<(NNODES / 16 + 3) / 4, 128, 0, stream>>>(mbuf, h, gwih, gwhh, gbih, gbhh);
  }

  // Set2Set, 3 processing steps
  k_fill<<<(BGR * 128 + 255) / 256, 256, 0, stream>>>(qstar, 0.f, BGR * 128);
  k_fill<<<(BGR * 64 + 255) / 256, 256, 0, stream>>>(hl, 0.f, BGR * 64);
  k_fill<<<(BGR * 64 + 255) / 256, 256, 0, stream>>>(clb, 0.f, BGR * 64);
  for (int it = 0; it < 3; it++) {
    k_lstm_gates<<<(BGR * 256 + 255) / 256, 256, 0, stream>>>(qstar, hl, lwih, lwhh,
                                                              lbih, lbhh, gates);
    k_lstm_update<<<(BGR * 64 + 255) / 256, 256, 0, stream>>>(gates, clb, hl);
    k_fill<<<1, 256, 0, stream>>>(emax, -INFINITY, BGR);
    k_fill<<<1, 256, 0, stream>>>(asum, 0.f, BGR);
    k_fill<<<(BGR * 64 + 255) / 256, 256, 0, stream>>>(rread, 0.f, BGR * 64);
    k_attn_e<<<(NNODES + 255) / 256, 256, 0, stream>>>(h, hl, batch, esc, emax);
    k_attn_a<<<(NNODES + 255) / 256, 256, 0, stream>>>(esc, emax, batch, asc, asum);
    k_readout<<<(NNODES * 64 + 255) / 256, 256, 0, stream>>>(asc, asum, h, batch, rread);
    k_qstar<<<(BGR * 128 + 255) / 256, 256, 0, stream>>>(hl, rread, qstar);
  }

  k_glin<<<(BGR * 64 + 255) / 256, 256, 0, stream>>>(qstar, l1w, l1b, gbuf);
  k_fill<<<1, 256, 0, stream>>>(refb, 0.f, BGR);
  k_ref<<<(NNODES + 255) / 256, 256, 0, stream>>>(z, batch, aref, eref, refb);
  k_energy<<<1, BGR, 0, stream>>>(gbuf, l2w, l2b, refb, out);
}